// InputSequenceNormalization_31121333026889
// MI455X (gfx1250) — compile-verified
//
#include <hip/hip_runtime.h>
#include <hip/hip_bf16.h>

// InputSequenceNormalization for MI455X (gfx1250).
// Memory-bound (AI ~0.6 flop/byte): goal is exactly one HBM read + one HBM
// write per element. Each workgroup owns one (b,f) row of 8000 fp32 values,
// stages it into LDS via the CDNA5 async global->LDS path (ASYNCcnt), then
// performs two reduction passes (mean, unbiased var) and one normalize pass
// entirely out of LDS.

constexpr int TPB   = 256;   // 8 wave32 per block
constexpr int T_LEN = 8000;  // frames per row (32 KB fp32)

typedef int v4i_t __attribute__((ext_vector_type(4)));

#if defined(__HIP_DEVICE_COMPILE__) && __has_builtin(__builtin_amdgcn_global_load_async_to_lds_b128)
#define USE_ASYNC_LDS 1
#endif

__device__ __forceinline__ float wave_reduce_add(float v) {
    // wave32 butterfly reduction
    #pragma unroll
    for (int off = 16; off > 0; off >>= 1)
        v += __shfl_xor(v, off, 32);
    return v;
}

__global__ __launch_bounds__(TPB)
void isn_kernel(const float* __restrict__ x,
                const float* __restrict__ lengths,
                float* __restrict__ out,
                int F) {
    __shared__ __align__(16) float row[T_LEN];
    __shared__ float red[16];

    const int r   = blockIdx.x;          // r = b*F + f
    const int b   = r / F;
    const int tid = threadIdx.x;
    const float* __restrict__ src = x   + (size_t)r * T_LEN;
    float*       __restrict__ dst = out + (size_t)r * T_LEN;

    // ---- stage the row into LDS (one HBM read) ----
#ifdef USE_ASYNC_LDS
    for (int c = tid; c < T_LEN / 4; c += TPB) {
        __builtin_amdgcn_global_load_async_to_lds_b128(
            (__attribute__((address_space(1))) v4i_t*)(src + 4 * c),
            (__attribute__((address_space(3))) v4i_t*)(&row[4 * c]),
            /*offset=*/0, /*cpol=*/0);
    }
    // each wave waits for its own async transfers, then block barrier
#if __has_builtin(__builtin_amdgcn_s_wait_asynccnt)
    __builtin_amdgcn_s_wait_asynccnt(0);
#else
    asm volatile("s_wait_asynccnt 0" ::: "memory");
#endif
#else
    for (int c = tid; c < T_LEN / 4; c += TPB) {
        float4 v = reinterpret_cast<const float4*>(src)[c];
        reinterpret_cast<float4*>(row)[c] = v;
    }
#endif
    __syncthreads();

    // valid frame count: jnp.round == round-half-even == rintf (RNE mode)
    const float nf = rintf(lengths[b] * (float)T_LEN);
    const int   n  = (int)nf;

    // ---- pass 1: masked sum -> mean ----
    float s = 0.0f;
    for (int c = tid; c < T_LEN / 4; c += TPB) {
        float4 v = reinterpret_cast<const float4*>(row)[c];
        const int t = 4 * c;
        if (t + 3 < n) {
            s += v.x + v.y + v.z + v.w;
        } else {
            if (t + 0 < n) s += v.x;
            if (t + 1 < n) s += v.y;
            if (t + 2 < n) s += v.z;
            if (t + 3 < n) s += v.w;
        }
    }
    s = wave_reduce_add(s);
    if ((tid & 31) == 0) red[tid >> 5] = s;
    __syncthreads();
    float tot = 0.0f;
    #pragma unroll
    for (int i = 0; i < TPB / 32; ++i) tot += red[i];
    const float mean = tot / nf;

    // ---- pass 2: masked sum of squared deviations -> unbiased var ----
    float q = 0.0f;
    for (int c = tid; c < T_LEN / 4; c += TPB) {
        float4 v = reinterpret_cast<const float4*>(row)[c];
        const int t = 4 * c;
        if (t + 3 < n) {
            float dx = v.x - mean; q += dx * dx;
            dx = v.y - mean;       q += dx * dx;
            dx = v.z - mean;       q += dx * dx;
            dx = v.w - mean;       q += dx * dx;
        } else {
            if (t + 0 < n) { float dx = v.x - mean; q += dx * dx; }
            if (t + 1 < n) { float dx = v.y - mean; q += dx * dx; }
            if (t + 2 < n) { float dx = v.z - mean; q += dx * dx; }
            if (t + 3 < n) { float dx = v.w - mean; q += dx * dx; }
        }
    }
    q = wave_reduce_add(q);
    if ((tid & 31) == 0) red[8 + (tid >> 5)] = q;
    __syncthreads();
    float qt = 0.0f;
    #pragma unroll
    for (int i = 0; i < TPB / 32; ++i) qt += red[8 + i];
    const float var = qt / (nf - 1.0f);
    const float inv = 1.0f / fmaxf(sqrtf(var), 1e-10f);

    // ---- pass 3: normalize ALL frames (one HBM write) ----
    for (int c = tid; c < T_LEN / 4; c += TPB) {
        float4 v = reinterpret_cast<const float4*>(row)[c];
        float4 o;
        o.x = (v.x - mean) * inv;
        o.y = (v.y - mean) * inv;
        o.z = (v.z - mean) * inv;
        o.w = (v.w - mean) * inv;
        reinterpret_cast<float4*>(dst)[c] = o;
    }
}

extern "C" void kernel_launch(void* const* d_in, const int* in_sizes, int n_in,
                              void* d_out, int out_size, void* d_ws, size_t ws_size,
                              hipStream_t stream) {
    (void)n_in; (void)d_ws; (void)ws_size; (void)out_size;
    const float* x       = (const float*)d_in[0];
    const float* lengths = (const float*)d_in[1];
    float*       out     = (float*)d_out;

    const int B    = in_sizes[1];                      // 64
    const int rows = in_sizes[0] / T_LEN;              // B*F = 5120
    const int F    = rows / B;                         // 80

    isn_kernel<<<rows, TPB, 0, stream>>>(x, lengths, out, F);
}